// VariationalAutoEncoder_88089779241018
// MI455X (gfx1250) — compile-verified
//
#include <hip/hip_runtime.h>

#define NN   50000
#define EE   1600000
#define BB   500
#define IN_  32
#define HH   128
#define LATD 64
#define DCC  8
#define NCC  7
#define HDD  256
#define NMAXD 100
#define PP   (NMAXD * (NMAXD - 1) / 2)   /* 4950 */
#define NPG  (NN / BB)                   /* 100 */
#define BN_EPS 1e-5f

typedef __attribute__((ext_vector_type(2))) float v2f;
typedef __attribute__((ext_vector_type(8))) float v8f;

// Full-precision CDNA5 matrix op: D(16x16 f32) = A(16x4 f32) x B(4x16 f32) + C
__device__ __forceinline__ v8f wmma_f32_16x16x4(v2f a, v2f b, v8f c) {
  return __builtin_amdgcn_wmma_f32_16x16x4_f32(false, a, false, b, (short)0, c,
                                               false, false);
}

// ---------------------------------------------------------------------------
// Edge scatter: agg[dst] += x[src].  One wave covers 128 contiguous floats.
// ---------------------------------------------------------------------------
template <int DIN>
__global__ void edge_scatter_kernel(const float* __restrict__ x,
                                    const int* __restrict__ src,
                                    const int* __restrict__ dst,
                                    float* __restrict__ agg) {
  const int CH = DIN / 4;
  long long t = (long long)blockIdx.x * blockDim.x + threadIdx.x;
  if (t >= (long long)EE * CH) return;
  int e = (int)(t / CH);
  int c = (int)(t % CH) * 4;
  int s = src[e], d = dst[e];
  const float4 v = *(const float4*)(x + (long long)s * DIN + c);
  float* p = agg + (long long)d * DIN + c;
  atomicAdd(p + 0, v.x);
  atomicAdd(p + 1, v.y);
  atomicAdd(p + 2, v.z);
  atomicAdd(p + 3, v.w);
}

// ---------------------------------------------------------------------------
// Fused GIN MLP: out = lrelu(W2 @ bn(lrelu(W1 @ (x+agg) + b1)) + b2)
// 16-row node tile per block; 8 waves each own a 16-column output tile.
// N % 16 == 0 -> no guards anywhere, EXEC stays all-ones.
// ---------------------------------------------------------------------------
template <int DIN>
__global__ __launch_bounds__(256) void gin_mlp_kernel(
    const float* __restrict__ x, const float* __restrict__ agg,
    const float* __restrict__ W1, const float* __restrict__ b1,
    const float* __restrict__ bng, const float* __restrict__ bnbe,
    const float* __restrict__ bnm, const float* __restrict__ bnv,
    const float* __restrict__ W2, const float* __restrict__ b2,
    float* __restrict__ out) {
  __shared__ float hs[16 * DIN];
  __shared__ float t1s[16 * HH];
  const int tid  = threadIdx.x;
  const int lane = tid & 31;
  const int wave = tid >> 5;  // 0..7 -> column tiles of H=128
  const long long row0 = (long long)blockIdx.x * 16;

  for (int i = tid; i < 16 * DIN; i += 256) {
    int r = i / DIN, c = i % DIN;
    long long g = (row0 + r) * DIN + c;
    hs[i] = x[g] + agg[g];
  }
  __syncthreads();

  const int m    = lane & 15;
  const int half = lane >> 4;
  const int col  = wave * 16 + (lane & 15);

  // GEMM1 over K = DIN, then bias + lrelu + BN into LDS
  v8f acc1 = {};
  {
    const float* Ap = hs + m * DIN + half * 2;          // 8B aligned
    const float* Bp = W1 + half * 2 * HH + col;
#pragma unroll
    for (int k0 = 0; k0 < DIN; k0 += 4) {
      v2f a = *(const v2f*)Ap;
      v2f b;
      b.x = Bp[0];
      b.y = Bp[HH];
      acc1 = wmma_f32_16x16x4(a, b, acc1);
      Ap += 4;
      Bp += 4 * HH;
    }
  }
  {
    float bi = b1[col];
    float sc = bng[col] * rsqrtf(bnv[col] + BN_EPS);
    float mm = bnm[col], be = bnbe[col];
    for (int r = 0; r < 8; ++r) {
      int rm = r + half * 8;
      float v = acc1[r] + bi;
      v = v > 0.f ? v : 0.2f * v;
      v = (v - mm) * sc + be;
      t1s[rm * HH + col] = v;
    }
  }
  __syncthreads();

  // GEMM2 over K = 128, then bias + lrelu to global
  v8f acc2 = {};
  {
    const float* Ap = t1s + m * HH + half * 2;
    const float* Bp = W2 + half * 2 * HH + col;
#pragma unroll
    for (int k0 = 0; k0 < HH; k0 += 4) {
      v2f a = *(const v2f*)Ap;
      v2f b;
      b.x = Bp[0];
      b.y = Bp[HH];
      acc2 = wmma_f32_16x16x4(a, b, acc2);
      Ap += 4;
      Bp += 4 * HH;
    }
  }
  {
    float bi = b2[col];
    for (int r = 0; r < 8; ++r) {
      int rm = r + half * 8;
      float v = acc2[r] + bi;
      v = v > 0.f ? v : 0.2f * v;
      out[(row0 + rm) * HH + col] = v;
    }
  }
}

// ---------------------------------------------------------------------------
// Generic WMMA GEMM: C[M,Nout] = act(A[M,K] @ W[K,Nout] + bias)
// One 16x16 tile per wave. Out-of-range rows/cols are CLAMPED (not masked):
// A row m only feeds D row m and B col n only feeds D col n, so duplicated
// edge rows/cols only corrupt D entries that are never stored. Loads are
// therefore unconditional: A pair -> global_load_b64, B -> 2x b32, no exec
// manipulation in the hot loop.  act: 0=none, 1=relu, 2=leaky_relu(0.2)
// ---------------------------------------------------------------------------
__global__ __launch_bounds__(128) void gemm_kernel(
    const float* __restrict__ A, int M, int K, int lda,
    const float* __restrict__ W, const float* __restrict__ bias,
    float* __restrict__ C, int Nout, int ldc, int act) {
  const int lane = threadIdx.x & 31;
  const int wave = threadIdx.x >> 5;
  const int tiles_n = (Nout + 15) >> 4;
  const int tiles_m = (M + 15) >> 4;
  long long t = (long long)blockIdx.x * 4 + wave;
  int tm = (int)(t / tiles_n);
  int tn = (int)(t % tiles_n);
  if (tm >= tiles_m) return;  // wave-uniform

  const int half = lane >> 4;
  const int m    = lane & 15;
  int arow = tm * 16 + m;
  arow = arow < M ? arow : (M - 1);        // clamp, see note above
  const int cn = tn * 16 + (lane & 15);
  const int cnc = cn < Nout ? cn : (Nout - 1);

  const float* Ap = A + (long long)arow * lda + half * 2;  // 8B aligned (lda even)
  const float* Bp = W + (long long)(half * 2) * Nout + cnc;
  const long long bstep = 4LL * Nout;

  v8f acc = {};
#pragma unroll 4
  for (int k0 = 0; k0 < K; k0 += 4) {
    v2f a = *(const v2f*)Ap;
    v2f b;
    b.x = Bp[0];
    b.y = Bp[Nout];
    acc = wmma_f32_16x16x4(a, b, acc);
    Ap += 4;
    Bp += bstep;
  }

  if (cn < Nout) {
    float bi = bias ? bias[cn] : 0.f;
    for (int r = 0; r < 8; ++r) {
      int rm = tm * 16 + r + half * 8;
      if (rm < M) {
        float v = acc[r] + bi;
        if (act == 1)      v = fmaxf(v, 0.f);
        else if (act == 2) v = v > 0.f ? v : 0.2f * v;
        C[(long long)rm * ldc + cn] = v;
      }
    }
  }
}

// ---------------------------------------------------------------------------
// global_add_pool over sorted batch (100 nodes/graph) fused with BN
// ---------------------------------------------------------------------------
__global__ void pool_bn_kernel(const float* __restrict__ x,
                               const float* __restrict__ g,
                               const float* __restrict__ be,
                               const float* __restrict__ mn,
                               const float* __restrict__ vr,
                               float* __restrict__ gbn) {
  int b = blockIdx.x, f = threadIdx.x;  // 500 x 128
  float s = 0.f;
  long long base = (long long)b * NPG;
  for (int i = 0; i < NPG; ++i) s += x[(base + i) * HH + f];
  gbn[b * HH + f] = (s - mn[f]) * rsqrtf(vr[f] + BN_EPS) * g[f] + be[f];
}

__global__ void bn_kernel(float* __restrict__ xio, const float* __restrict__ g,
                          const float* __restrict__ be,
                          const float* __restrict__ mn,
                          const float* __restrict__ vr, int rows, int cols) {
  long long t = (long long)blockIdx.x * blockDim.x + threadIdx.x;
  if (t >= (long long)rows * cols) return;
  int f = (int)(t % cols);
  xio[t] = (xio[t] - mn[f]) * rsqrtf(vr[f] + BN_EPS) * g[f] + be[f];
}

// cond = lin2(relu(lin1(stats))) written into hcat[:, 128:136]
__global__ void cond_kernel(const float* __restrict__ stats,
                            const float* __restrict__ W1,
                            const float* __restrict__ b1,
                            const float* __restrict__ W2,
                            const float* __restrict__ b2,
                            float* __restrict__ out) {
  int b = blockIdx.x * blockDim.x + threadIdx.x;
  if (b >= BB) return;
  float tmp[DCC];
  for (int j = 0; j < DCC; ++j) {
    float s = b1[j];
    for (int i = 0; i < NCC; ++i) s += stats[b * NCC + i] * W1[i * DCC + j];
    tmp[j] = fmaxf(s, 0.f);
  }
  for (int j = 0; j < DCC; ++j) {
    float s = b2[j];
    for (int i = 0; i < DCC; ++i) s += tmp[i] * W2[i * DCC + j];
    out[(long long)b * (HH + DCC) + j] = s;
  }
}

// gumbel-softmax(hard) -> symmetric adjacency; each output element once
__global__ void adj_kernel(const float* __restrict__ logits,
                           const float* __restrict__ gum,
                           float* __restrict__ out) {
  long long t = (long long)blockIdx.x * blockDim.x + threadIdx.x;
  if (t >= (long long)BB * NMAXD * NMAXD) return;
  int b  = (int)(t / (NMAXD * NMAXD));
  int rc = (int)(t % (NMAXD * NMAXD));
  int r = rc / NMAXD, c = rc % NMAXD;
  if (r == c) { out[t] = 0.f; return; }
  int i = r < c ? r : c;
  int j = r < c ? c : r;
  int p = i * (2 * NMAXD - i - 1) / 2 + (j - i - 1);
  long long lb = (long long)b * (2 * PP) + 2 * p;
  long long gb = ((long long)b * PP + p) * 2;
  float a0 = logits[lb]     + gum[gb];
  float a1 = logits[lb + 1] + gum[gb + 1];
  float mx = fmaxf(a0, a1);
  float e0 = expf(a0 - mx), e1 = expf(a1 - mx);
  float inv = 1.f / (e0 + e1);
  float y0 = e0 * inv, y1 = e1 * inv;
  float hard = (y0 >= y1) ? 1.f : 0.f;
  float e = hard + y0;  // replicate reference fp arithmetic
  e = e - y0;
  out[t] = e;
}

// ---------------------------------------------------------------------------
extern "C" void kernel_launch(void* const* d_in, const int* in_sizes, int n_in,
                              void* d_out, int out_size, void* d_ws,
                              size_t ws_size, hipStream_t stream) {
  (void)in_sizes; (void)n_in; (void)out_size; (void)ws_size;
  auto F = [&](int i) { return (const float*)d_in[i]; };

  const float* x0    = F(0);
  const float* stats = F(1);
  const float* gum   = F(2);
  // params, jax tree-flatten (alphabetical) order
  const float *bn_be = F(3), *bn_g = F(4), *bn_m = F(5), *bn_v = F(6);
  const float *c1W = F(7), *c1b = F(8), *c2W = F(9), *c2b = F(10);
  // convs[k] at 11 + 8k: bn.be, bn.g, bn.m, bn.v, lin1.W, lin1.b, lin2.W, lin2.b
  const float *decW0 = F(35), *decb0 = F(36);
  const float *decW1 = F(37), *decb1 = F(38);
  const float *decW2 = F(39), *decb2 = F(40);
  const float *fcW = F(41), *fcb = F(42);
  const float *fmW = F(43), *fmb = F(44);
  const float *fl1W = F(45), *fl1b = F(46);
  const float *fl2W = F(47), *fl2b = F(48);
  const float *flbn_be = F(49), *flbn_g = F(50), *flbn_m = F(51), *flbn_v = F(52);
  const int* edge = (const int*)d_in[53];
  const int* esrc = edge;
  const int* edst = edge + EE;

  float* ws  = (float*)d_ws;
  float* xA  = ws;
  float* xB  = ws + (long long)NN * HH;
  float* agg = ws + 2LL * NN * HH;
  // small head buffers live inside agg region (graph phase done by then)
  float* gbn   = agg;
  float* hcat  = agg + 100000;   // [500,136]
  float* t1    = agg + 200000;   // [500,128]
  float* t2    = agg + 300000;   // [500,128]
  float* z     = agg + 400000;   // [500,64]
  float* d1    = agg + 500000;   // [500,256]
  float* d2    = agg + 700000;   // [500,256]
  float* logit = agg + 900000;   // [500,9900]
  float* outp  = (float*)d_out;

  auto gemm = [&](const float* A, int M, int K, int lda, const float* W,
                  const float* bias, float* C, int Nout, int ldc, int act) {
    int tiles = ((M + 15) / 16) * ((Nout + 15) / 16);
    gemm_kernel<<<(tiles + 3) / 4, 128, 0, stream>>>(A, M, K, lda, W, bias, C,
                                                     Nout, ldc, act);
  };

  // ---- GIN layer 0 (din = 32) ----
  hipMemsetAsync(agg, 0, (size_t)NN * IN_ * sizeof(float), stream);
  {
    long long tt = (long long)EE * (IN_ / 4);
    edge_scatter_kernel<IN_><<<(int)((tt + 255) / 256), 256, 0, stream>>>(
        x0, esrc, edst, agg);
    gin_mlp_kernel<IN_><<<NN / 16, 256, 0, stream>>>(
        x0, agg, F(15), F(16), F(12), F(11), F(13), F(14), F(17), F(18), xA);
  }
  // ---- GIN layer 1 (din = 128) ----
  hipMemsetAsync(agg, 0, (size_t)NN * HH * sizeof(float), stream);
  {
    long long tt = (long long)EE * (HH / 4);
    edge_scatter_kernel<HH><<<(int)((tt + 255) / 256), 256, 0, stream>>>(
        xA, esrc, edst, agg);
    gin_mlp_kernel<HH><<<NN / 16, 256, 0, stream>>>(
        xA, agg, F(23), F(24), F(20), F(19), F(21), F(22), F(25), F(26), xB);
  }
  // ---- GIN layer 2 (din = 128) ----
  hipMemsetAsync(agg, 0, (size_t)NN * HH * sizeof(float), stream);
  {
    long long tt = (long long)EE * (HH / 4);
    edge_scatter_kernel<HH><<<(int)((tt + 255) / 256), 256, 0, stream>>>(
        xB, esrc, edst, agg);
    gin_mlp_kernel<HH><<<NN / 16, 256, 0, stream>>>(
        xB, agg, F(31), F(32), F(28), F(27), F(29), F(30), F(33), F(34), xA);
  }

  // ---- pool + bn, head ----
  pool_bn_kernel<<<BB, HH, 0, stream>>>(xA, bn_g, bn_be, bn_m, bn_v, gbn);
  gemm(gbn, BB, HH, HH, fcW, fcb, hcat, HH, HH + DCC, 0);           // fc
  cond_kernel<<<(BB + 127) / 128, 128, 0, stream>>>(stats, c1W, c1b, c2W, c2b,
                                                    hcat + HH);
  gemm(hcat, BB, HH + DCC, HH + DCC, fl1W, fl1b, t1, HH, HH, 2);    // fl1+lrelu
  bn_kernel<<<(BB * HH + 255) / 256, 256, 0, stream>>>(t1, flbn_g, flbn_be,
                                                       flbn_m, flbn_v, BB, HH);
  gemm(t1, BB, HH, HH, fl2W, fl2b, t2, HH, HH, 0);                  // fl2
  gemm(t2, BB, HH, HH, fmW, fmb, z, LATD, LATD, 0);                 // fc_mu
  gemm(z, BB, LATD, LATD, decW0, decb0, d1, HDD, HDD, 1);           // dec0+relu
  gemm(d1, BB, HDD, HDD, decW1, decb1, d2, HDD, HDD, 1);            // dec1+relu
  gemm(d2, BB, HDD, HDD, decW2, decb2, logit, 2 * PP, 2 * PP, 0);   // dec2

  adj_kernel<<<(BB * NMAXD * NMAXD + 255) / 256, 256, 0, stream>>>(logit, gum,
                                                                   outp);
}